// MoELayer_62483184222256
// MI455X (gfx1250) — compile-verified
//
#include <hip/hip_runtime.h>
#include <hip/hip_bf16.h>

// MoE layer, restructured: out[b,o] = sum_e (sum_s w[b,s,e]*x[b,s,i]) * EW[e,o,i]
//                                   + sum_e (sum_s w[b,s,e]) * EB[e,o]
// B=8, S=2048, I=1024, O=1024, E=8, K=2  (fp32 throughout)
//
// HBM traffic: x read once (64 MB, fused gate+reduce through LDS),
// expert_w read once (32 MB), everything else is KB-scale -> ~100 MB total
// => ~4.3 us at 23.3 TB/s. Compute is trivial after the algebraic refactor.

#define Bsz 8
#define Ssz 2048
#define Isz 1024
#define Osz 1024
#define Esz 8

#define TOK_PER_BLK 32                    // tokens staged in LDS per block
#define LDS_X_FLOATS (TOK_PER_BLK * Isz)  // 32768 floats = 128 KB
#define LDS_W_FLOATS (TOK_PER_BLK * Esz)  // 256 floats

typedef __attribute__((ext_vector_type(2))) float v2f;
typedef __attribute__((ext_vector_type(8))) float v8f;

// ---------------------------------------------------------------- zero fill
__global__ __launch_bounds__(256) void zero_f32(float* __restrict__ p, int n) {
  int i = blockIdx.x * 256 + threadIdx.x;
  if (i < n) p[i] = 0.0f;
}

// ---------------------------------------------------------------- fused kernel
// Per block: 32 tokens of one batch row.
//  P1: global->LDS stage of x rows (x touched exactly once from HBM)
//  P2: gating per token (8 dots from LDS, wave reduce, softmax/top2/renorm)
//  P3: z[b,e,i] partial accumulation from LDS, atomics into global z
__global__ __launch_bounds__(256) void moe_gate_accum(const float* __restrict__ x,
                                                      const float* __restrict__ gw,
                                                      const float* __restrict__ gb,
                                                      float* __restrict__ z,
                                                      float* __restrict__ wsum) {
  extern __shared__ float lds[];
  float* xs  = lds;                  // [TOK_PER_BLK][Isz]
  float* wsm = lds + LDS_X_FLOATS;   // [TOK_PER_BLK][Esz]

  const int tid  = threadIdx.x;
  const int b    = blockIdx.x >> 6;            // 64 s-chunks per batch row
  const int s0   = (blockIdx.x & 63) * TOK_PER_BLK;
  const float* xblk = x + ((size_t)b * Ssz + s0) * Isz;

  // ---- P1: coalesced float4 stage, 32768 floats / 256 threads = 32 iters
#pragma unroll 4
  for (int it = 0; it < LDS_X_FLOATS / (256 * 4); ++it) {
    const int f = (it * 256 + tid) * 4;
    *(float4*)(xs + f) = *(const float4*)(xblk + f);
  }
  __syncthreads();

  // ---- P2: gating, one wave per 4 tokens
  const int wave = tid >> 5;
  const int lane = tid & 31;
  for (int tl = wave * 4; tl < wave * 4 + 4; ++tl) {
    float acc[Esz];
#pragma unroll
    for (int e = 0; e < Esz; ++e) acc[e] = 0.0f;
#pragma unroll
    for (int it = 0; it < Isz / 128; ++it) {   // 32 lanes * float4
      const int i = it * 128 + lane * 4;
      const float4 xv = *(const float4*)(xs + tl * Isz + i);
#pragma unroll
      for (int e = 0; e < Esz; ++e) {
        const float4 gv = *(const float4*)(gw + e * Isz + i);
        acc[e] = fmaf(xv.x, gv.x, acc[e]);
        acc[e] = fmaf(xv.y, gv.y, acc[e]);
        acc[e] = fmaf(xv.z, gv.z, acc[e]);
        acc[e] = fmaf(xv.w, gv.w, acc[e]);
      }
    }
#pragma unroll
    for (int off = 16; off >= 1; off >>= 1)
#pragma unroll
      for (int e = 0; e < Esz; ++e)
        acc[e] += __shfl_xor(acc[e], off, 32);

    if (lane == 0) {
      float l[Esz], p[Esz];
      float m = -1e30f;
#pragma unroll
      for (int e = 0; e < Esz; ++e) { l[e] = acc[e] + gb[e]; m = fmaxf(m, l[e]); }
#pragma unroll
      for (int e = 0; e < Esz; ++e) p[e] = expf(l[e] - m);  // softmax Z cancels
      int i0 = 0;
#pragma unroll
      for (int e = 1; e < Esz; ++e) if (p[e] > p[i0]) i0 = e;
      int i1 = (i0 == 0) ? 1 : 0;
#pragma unroll
      for (int e = 0; e < Esz; ++e) if (e != i0 && p[e] > p[i1]) i1 = e;
      const float s  = p[i0] + p[i1];
      const float g0 = p[i0] / s;
      const float g1 = p[i1] / s;
#pragma unroll
      for (int e = 0; e < Esz; ++e)
        wsm[tl * Esz + e] = (e == i0) ? g0 : ((e == i1) ? g1 : 0.0f);
    }
  }
  __syncthreads();

  // ---- P3: each thread owns 4 contiguous i, accumulate over 32 tokens
  const int i0 = tid * 4;
  float acc[Esz][4];
#pragma unroll
  for (int e = 0; e < Esz; ++e)
#pragma unroll
    for (int j = 0; j < 4; ++j) acc[e][j] = 0.0f;

  for (int s = 0; s < TOK_PER_BLK; ++s) {
    const float4 xv = *(const float4*)(xs + s * Isz + i0);
    const float* wr = wsm + s * Esz;             // uniform -> LDS broadcast
#pragma unroll
    for (int e = 0; e < Esz; ++e) {
      const float wv = wr[e];
      acc[e][0] = fmaf(wv, xv.x, acc[e][0]);
      acc[e][1] = fmaf(wv, xv.y, acc[e][1]);
      acc[e][2] = fmaf(wv, xv.z, acc[e][2]);
      acc[e][3] = fmaf(wv, xv.w, acc[e][3]);
    }
  }
#pragma unroll
  for (int e = 0; e < Esz; ++e)
#pragma unroll
    for (int j = 0; j < 4; ++j)
      atomicAdd(&z[(size_t)(b * Esz + e) * Isz + i0 + j], acc[e][j]);

  // ---- per-block wsum partial (bias term)
  if (tid < Esz) {
    float sw = 0.0f;
    for (int s = 0; s < TOK_PER_BLK; ++s) sw += wsm[s * Esz + tid];
    atomicAdd(&wsum[b * Esz + tid], sw);
  }
}

// ---------------------------------------------------------------- WMMA GEMM
// out[16(pad from 8), 16-wide O tile] += z_e(16x1024) @ EW_e(16x1024)^T via
// V_WMMA_F32_16X16X4_F32, K looped over I in steps of 4. One wave per block.
// z is padded to 16 rows (rows 8..15 stay zero) so A loads are branch-free
// and EXEC stays all-1s around every WMMA.
__global__ __launch_bounds__(32) void moe_out(const float* __restrict__ z,
                                              const float* __restrict__ ew,
                                              const float* __restrict__ eb,
                                              const float* __restrict__ wsum,
                                              float* __restrict__ out) {
  const int e    = blockIdx.x & 7;               // expert (K-partial)
  const int o0   = ((blockIdx.x >> 3) << 4);     // 16-wide O tile
  const int lane = threadIdx.x;
  const int n    = lane & 15;                    // A: row M=n / B,C: col N=n
  const int kh   = lane >> 4;                    // 0: K=0,1  1: K=2,3 (ISA A layout)

  // A[m][k] = z[m, e, i+k]  (m = n, rows 8..15 are the zero pad)
  const float* ap = z  + ((size_t)(n * Esz + e)) * Isz + 2 * kh;
  // B[k][n] = ew[e, o0+n, i+k]
  const float* bp = ew + ((size_t)(e * Osz + o0 + n)) * Isz + 2 * kh;

  v8f c = {};
  for (int i = 0; i < Isz; i += 4) {
    if ((i & 255) == 0 && (i + 256) < Isz)
      __builtin_prefetch(bp + i + 256, 0, 0);    // global_prefetch_b8 on HBM stream
    const v2f a = *(const v2f*)(ap + i);
    const v2f b = *(const v2f*)(bp + i);
    c = __builtin_amdgcn_wmma_f32_16x16x4_f32(
        /*neg_a=*/false, a, /*neg_b=*/false, b,
        /*c_mod=*/(short)0, c, /*reuse_a=*/false, /*reuse_b=*/false);
  }

  // C/D layout: VGPR r, lanes 0-15 -> M=r (real rows), lanes 16-31 -> M=8+r (pad)
  if (kh == 0) {
#pragma unroll
    for (int r = 0; r < Bsz; ++r) {
      float val = c[r];
      if (e == 0) {                              // add bias term exactly once
        float bb = 0.0f;
#pragma unroll
        for (int ee = 0; ee < Esz; ++ee)
          bb = fmaf(wsum[r * Esz + ee], eb[ee * Osz + o0 + n], bb);
        val += bb;
      }
      atomicAdd(out + r * Osz + o0 + n, val);
    }
  }
}

// ---------------------------------------------------------------- launch
extern "C" void kernel_launch(void* const* d_in, const int* in_sizes, int n_in,
                              void* d_out, int out_size, void* d_ws, size_t ws_size,
                              hipStream_t stream) {
  const float* x  = (const float*)d_in[0];   // [B,S,I]
  const float* gw = (const float*)d_in[1];   // [E,I]
  const float* gb = (const float*)d_in[2];   // [E]
  const float* ew = (const float*)d_in[3];   // [E,O,I]
  const float* eb = (const float*)d_in[4];   // [E,O]
  float* out = (float*)d_out;                // [B,O] = 8192 floats
  float* ws  = (float*)d_ws;

  float* z    = ws;                 // [16(pad), E, I] = 131072 floats
  float* wsum = ws + 131072;        // [B, E]          = 64 floats

  // zero z + wsum (contiguous) and out every call (graph-replay safe)
  zero_f32<<<(131072 + 64 + 255) / 256, 256, 0, stream>>>(z, 131072 + 64);
  zero_f32<<<(Bsz * Osz + 255) / 256, 256, 0, stream>>>(out, Bsz * Osz);

  const size_t lds_bytes = (LDS_X_FLOATS + LDS_W_FLOATS) * sizeof(float); // ~129 KB
  moe_gate_accum<<<(Bsz * Ssz) / TOK_PER_BLK, 256, lds_bytes, stream>>>(
      x, gw, gb, z, wsum);

  moe_out<<<(Osz / 16) * Esz, 32, 0, stream>>>(z, ew, eb, wsum, out);
}